// GraphAdjNorm_35759897706670
// MI455X (gfx1250) — compile-verified
//
#include <hip/hip_runtime.h>
#include <cstdint>
#include <cstddef>

// GraphAdjNorm: out[b,i,j] = x[b,i,j] * rsqrt(d_i) * rsqrt(d_j), d = rowsum(x).
// Bandwidth-bound (768 MiB min traffic @ 23.3 TB/s ~= 33 us). Two passes:
//   pass 1: wave-per-row degree reduction -> inv_sqrt in d_ws
//   pass 2: elementwise scale; inv_sqrt[b,:] staged in LDS via CDNA5
//           global_load_async_to_lds_b128 (+ s_wait_asynccnt), NT stores,
//           reversed matrix order to harvest L2 residue of pass 1.

typedef float v4f __attribute__((ext_vector_type(4)));

#define NCOLS 2048

// ---- CDNA5 async global->LDS copy (ISA 15.18 / CDNA5 ISA) ----
// lds_off: byte offset within the wave's LDS allocation (low 32 bits of a
// generic pointer to a __shared__ object). gaddr: 64-bit global address.
__device__ __forceinline__ void async_b128_to_lds(uint32_t lds_off, const void* gaddr) {
    asm volatile("global_load_async_to_lds_b128 %0, %1, off"
                 :
                 : "v"(lds_off), "v"((uint64_t)gaddr)
                 : "memory");
}

__device__ __forceinline__ void wait_asynccnt0() {
#if defined(__has_builtin)
#if __has_builtin(__builtin_amdgcn_s_wait_asynccnt)
    __builtin_amdgcn_s_wait_asynccnt(0);
#else
    asm volatile("s_wait_asynccnt 0" ::: "memory");
#endif
#else
    asm volatile("s_wait_asynccnt 0" ::: "memory");
#endif
}

// ---------------- Pass 1: inv_sqrt of row degrees ----------------
// One wave32 per row; 8 rows per 256-thread block.
__global__ void __launch_bounds__(256)
row_inv_sqrt_kernel(const float* __restrict__ x, float* __restrict__ inv, int nrows) {
    const int lane = threadIdx.x & 31;
    const int wave = threadIdx.x >> 5;
    const int row  = blockIdx.x * 8 + wave;
    if (row >= nrows) return;

    const v4f* __restrict__ xr = (const v4f*)(x + (size_t)row * NCOLS);
    float s = 0.0f;
#pragma unroll
    for (int c = 0; c < NCOLS / 4 / 32; ++c) {        // 16 x global_load_b128
        v4f v = xr[c * 32 + lane];
        s += v.x + v.y + v.z + v.w;
    }
    // wave32 butterfly reduction
#pragma unroll
    for (int off = 16; off > 0; off >>= 1)
        s += __shfl_xor(s, off, 32);

    if (lane == 0) inv[row] = rsqrtf(s);
}

// ---------------- Pass 2: out = x * is_i * is_j ----------------
// Block owns 8 rows of one matrix; inv_sqrt[b,0:2048] staged in LDS (8 KB)
// via async-to-LDS. Blocks iterate matrices in REVERSE so the matrices still
// resident in the 192 MB L2 after pass 1 are consumed first.
__global__ void __launch_bounds__(256)
scale_kernel(const float* __restrict__ x, const float* __restrict__ inv,
             float* __restrict__ out) {
    constexpr int ROWS   = 8;
    constexpr int CHUNKS = NCOLS / ROWS;              // row-chunks per matrix

    __shared__ float sj[NCOLS];                       // 8 KB

    const int blk = (int)(gridDim.x - 1u - blockIdx.x);
    const int b   = blk / CHUNKS;
    const int rc  = blk % CHUNKS;
    const int t   = threadIdx.x;

    // Cooperative async copy: 512 x v4f, 2 per thread.
    const v4f* __restrict__ invb4 = (const v4f*)(inv + (size_t)b * NCOLS);
#pragma unroll
    for (int k = 0; k < NCOLS / 4 / 256; ++k) {       // 2 iterations
        const int idx = t + k * 256;
        const uint32_t loff = (uint32_t)(uintptr_t)(&sj[idx * 4]);
        async_b128_to_lds(loff, (const void*)(invb4 + idx));
    }
    wait_asynccnt0();
    __syncthreads();

    const int row0 = rc * ROWS;
    const v4f* __restrict__ xb  = (const v4f*)(x   + ((size_t)b * NCOLS + row0) * NCOLS);
    v4f* __restrict__       ob  = (v4f*)      (out + ((size_t)b * NCOLS + row0) * NCOLS);
    const v4f* __restrict__ sj4 = (const v4f*)sj;

#pragma unroll
    for (int r = 0; r < ROWS; ++r) {
        const float si = sj[row0 + r];
        const v4f* __restrict__ xr = xb + (size_t)r * (NCOLS / 4);
        v4f* __restrict__       orow = ob + (size_t)r * (NCOLS / 4);
#pragma unroll
        for (int k = 0; k < NCOLS / 4 / 256; ++k) {   // 2 iterations
            const int c = t + k * 256;
            v4f xv = __builtin_nontemporal_load(xr + c);   // streamed, dead after
            v4f jv = sj4[c];                               // ds_load_b128, conflict-free
            v4f o  = xv * (jv * si);
            __builtin_nontemporal_store(o, orow + c);      // NT: don't pollute L2
        }
    }
}

extern "C" void kernel_launch(void* const* d_in, const int* in_sizes, int n_in,
                              void* d_out, int out_size, void* d_ws, size_t ws_size,
                              hipStream_t stream) {
    const float* x   = (const float*)d_in[0];
    float*       out = (float*)d_out;
    float*       inv = (float*)d_ws;                  // B*N floats = 128 KB

    const long long total = (long long)in_sizes[0];   // B * N * N
    const int nrows = (int)(total / NCOLS);           // B * N
    const int B     = nrows / NCOLS;

    // Pass 1: one wave per row, 8 rows per block.
    row_inv_sqrt_kernel<<<(nrows + 7) / 8, 256, 0, stream>>>(x, inv, nrows);

    // Pass 2: B * (N/8) blocks, 8 rows each.
    const int chunks = NCOLS / 8;
    scale_kernel<<<B * chunks, 256, 0, stream>>>(x, inv, out);
}